// IdiomExtractor_36155034698433
// MI455X (gfx1250) — compile-verified
//
#include <hip/hip_runtime.h>

#define Bn 64
#define Tn 512
#define Hn 768
#define Cn 3
#define NROWS (Bn * Tn)        // 32768
#define EPSf 1e-5f
#define NEGF (-1e30f)

typedef float v2f __attribute__((ext_vector_type(2)));
typedef float v8f __attribute__((ext_vector_type(8)));

// ---- workspace layout (floats) ----
// [0,768)       colsum
// [768,1536)    colsumsq
// [1536,2304)   scale
// [2304,2320)   bias' (padded to 16)
// [2320,2320+16*768)  WpT[16][768]  (BN-folded, N-padded weights, row n, col k)
#define WS_SUM    0
#define WS_SUMSQ  768
#define WS_SCALE  1536
#define WS_BP     2304
#define WS_WPT    2320

__device__ __forceinline__ float lse3(float a, float b, float c) {
    float m = fmaxf(a, fmaxf(b, c));
    return m + __logf(__expf(a - m) + __expf(b - m) + __expf(c - m));
}

// ------------------------------------------------------------------
// K0: zero stats accumulators and the log-likelihood output slot
// ------------------------------------------------------------------
__global__ void k_init(float* ws, float* out0) {
    for (int i = threadIdx.x; i < 1536; i += 256) ws[i] = 0.0f;
    if (threadIdx.x == 0) out0[0] = 0.0f;
}

// ------------------------------------------------------------------
// K1: column sum / sum-of-squares over the 32768x768 matrix.
// 256 blocks x 256 threads; block handles 128 rows; thread t handles
// columns {t, t+256, t+512} -> fully coalesced loads.
// ------------------------------------------------------------------
__global__ void k_stats(const float* __restrict__ x, float* __restrict__ ws) {
    int t = threadIdx.x;
    const float* p = x + (size_t)blockIdx.x * 128 * Hn;
    float s0 = 0.f, s1 = 0.f, s2 = 0.f, q0 = 0.f, q1 = 0.f, q2 = 0.f;
    for (int r = 0; r < 128; ++r) {
        float a = p[t], b = p[t + 256], c = p[t + 512];
        s0 += a; q0 += a * a;
        s1 += b; q1 += b * b;
        s2 += c; q2 += c * c;
        p += Hn;
    }
    unsafeAtomicAdd(&ws[WS_SUM + t], s0);
    unsafeAtomicAdd(&ws[WS_SUM + t + 256], s1);
    unsafeAtomicAdd(&ws[WS_SUM + t + 512], s2);
    unsafeAtomicAdd(&ws[WS_SUMSQ + t], q0);
    unsafeAtomicAdd(&ws[WS_SUMSQ + t + 256], q1);
    unsafeAtomicAdd(&ws[WS_SUMSQ + t + 512], q2);
}

// ------------------------------------------------------------------
// K2: fold BatchNorm into the classifier:
//   scale = gamma * rsqrt(var+eps); shift = beta - mu*scale
//   WpT[n][k] = W[n][k]*scale[k] (n<3, else 0);  bp[n] = b[n]+sum_k shift[k]W[n][k]
// Single block, 256 threads.
// ------------------------------------------------------------------
__global__ void k_prep(const float* __restrict__ gamma, const float* __restrict__ beta,
                       const float* __restrict__ W, const float* __restrict__ bvec,
                       float* __restrict__ ws) {
    int tid = threadIdx.x;
    float part0 = 0.f, part1 = 0.f, part2 = 0.f;
    const float inv_n = 1.0f / (float)NROWS;
    for (int k = tid; k < Hn; k += 256) {
        float mu  = ws[WS_SUM + k] * inv_n;
        float var = ws[WS_SUMSQ + k] * inv_n - mu * mu;
        float sc  = gamma[k] * rsqrtf(var + EPSf);
        float sh  = beta[k] - mu * sc;
        ws[WS_SCALE + k] = sc;
        part0 += sh * W[0 * Hn + k];
        part1 += sh * W[1 * Hn + k];
        part2 += sh * W[2 * Hn + k];
    }
    __shared__ float red[3][256];
    red[0][tid] = part0; red[1][tid] = part1; red[2][tid] = part2;
    __syncthreads();
    for (int s = 128; s > 0; s >>= 1) {
        if (tid < s) {
            red[0][tid] += red[0][tid + s];
            red[1][tid] += red[1][tid + s];
            red[2][tid] += red[2][tid + s];
        }
        __syncthreads();
    }
    if (tid < 16) ws[WS_BP + tid] = (tid < Cn) ? (bvec[tid] + red[tid][0]) : 0.0f;
    __syncthreads();
    for (int idx = tid; idx < 16 * Hn; idx += 256) {
        int n = idx / Hn, k = idx - n * Hn;
        ws[WS_WPT + idx] = (n < Cn) ? W[n * Hn + k] * ws[WS_SCALE + k] : 0.0f;
    }
}

// ------------------------------------------------------------------
// K3: O = x @ WpT' + bp via v_wmma_f32_16x16x4_f32.
// One wave per 16-row tile (2048 tiles), 8 waves per block.
// A 16x4 f32 layout: lanes0-15 M=0..15 {K0,K1}; lanes16-31 {K2,K3}.
// B 4x16 f32 layout: VGPR0 lanes0-15=(K0,N), lanes16-31=(K2,N); VGPR1=(K1/K3).
// C/D: VGPR j = (M = half*8 + j, N = lane&15).
// ------------------------------------------------------------------
__global__ void k_gemm(const float* __restrict__ x, const float* __restrict__ ws,
                       float* __restrict__ O) {
    int wave = threadIdx.x >> 5;
    int lane = threadIdx.x & 31;
    int half = lane >> 4;
    int mrow = lane & 15;
    int tile = blockIdx.x * 8 + wave;

    const float* WpT = ws + WS_WPT;
    const float* bp  = ws + WS_BP;

    const float* arow = x + (size_t)(tile * 16 + mrow) * Hn + 2 * half;
    const float* brow = WpT + (size_t)mrow * Hn + 2 * half;

    float bb = bp[mrow];
    v8f c;
    #pragma unroll
    for (int i = 0; i < 8; ++i) c[i] = bb;

    #pragma unroll 4
    for (int k = 0; k < Hn; k += 4) {
        v2f a = *(const v2f*)(arow + k);
        v2f b = *(const v2f*)(brow + k);
        c = __builtin_amdgcn_wmma_f32_16x16x4_f32(false, a, false, b,
                                                  (short)0, c, false, false);
    }

    if (mrow < Cn) {
        int rbase = tile * 16 + half * 8;
        #pragma unroll
        for (int j = 0; j < 8; ++j)
            O[(size_t)(rbase + j) * Cn + mrow] = c[j];
    }
}

// ------------------------------------------------------------------
// K4: CRF numerator + forward-algorithm denominator.
// The masked forward step is a log-semiring 3x3 matmul (associative);
// masked steps are the identity. We only need the final alpha, so do a
// parallel tree REDUCTION over the 511 per-step matrices.
// 64 blocks (one per batch) x 128 threads (4 timesteps each).
// ------------------------------------------------------------------
__global__ void k_crf(const float* __restrict__ Omat, const int* __restrict__ labels,
                      const float* __restrict__ st, const float* __restrict__ et,
                      const float* __restrict__ tr, float* __restrict__ out0) {
    int b = blockIdx.x, tid = threadIdx.x;
    const int* lab = labels + b * Tn;
    const float* Ob = Omat + (size_t)b * Tn * Cn;

    float t00 = tr[0], t01 = tr[1], t02 = tr[2];
    float t10 = tr[3], t11 = tr[4], t12 = tr[5];
    float t20 = tr[6], t21 = tr[7], t22 = tr[8];

    // chunk product, init = identity in the log semiring
    float P00 = 0.f,  P01 = NEGF, P02 = NEGF;
    float P10 = NEGF, P11 = 0.f,  P12 = NEGF;
    float P20 = NEGF, P21 = NEGF, P22 = 0.f;
    float pnum = 0.f;
    int   pcnt = 0;

    int tbase = 1 + tid * 4;
    for (int u = 0; u < 4; ++u) {
        int t = tbase + u;
        if (t < Tn) {
            int lt = lab[t];
            if (lt != 0) {          // mask[t]: identity when masked -> skip
                ++pcnt;
                int lp = lab[t - 1];
                const float* e = Ob + t * Cn;
                float e0 = e[0], e1 = e[1], e2 = e[2];
                pnum += tr[lp * Cn + lt] + e[lt];
                float M00 = t00 + e0, M01 = t01 + e1, M02 = t02 + e2;
                float M10 = t10 + e0, M11 = t11 + e1, M12 = t12 + e2;
                float M20 = t20 + e0, M21 = t21 + e1, M22 = t22 + e2;
                float n00 = lse3(P00 + M00, P01 + M10, P02 + M20);
                float n01 = lse3(P00 + M01, P01 + M11, P02 + M21);
                float n02 = lse3(P00 + M02, P01 + M12, P02 + M22);
                float n10 = lse3(P10 + M00, P11 + M10, P12 + M20);
                float n11 = lse3(P10 + M01, P11 + M11, P12 + M21);
                float n12 = lse3(P10 + M02, P11 + M12, P12 + M22);
                float n20 = lse3(P20 + M00, P21 + M10, P22 + M20);
                float n21 = lse3(P20 + M01, P21 + M11, P22 + M21);
                float n22 = lse3(P20 + M02, P21 + M12, P22 + M22);
                P00 = n00; P01 = n01; P02 = n02;
                P10 = n10; P11 = n11; P12 = n12;
                P20 = n20; P21 = n21; P22 = n22;
            }
        }
    }

    __shared__ float SP[128][9];
    __shared__ float Snum[128];
    __shared__ int   Scnt[128];
    SP[tid][0] = P00; SP[tid][1] = P01; SP[tid][2] = P02;
    SP[tid][3] = P10; SP[tid][4] = P11; SP[tid][5] = P12;
    SP[tid][6] = P20; SP[tid][7] = P21; SP[tid][8] = P22;
    Snum[tid] = pnum; Scnt[tid] = pcnt;
    __syncthreads();

    // ordered tree reduction: SP[tid] (earlier t) (x) SP[tid+s] (later t)
    for (int s = 1; s < 128; s <<= 1) {
        if ((tid & (2 * s - 1)) == 0) {
            float A[9], Bm[9], Cc[9];
            #pragma unroll
            for (int i = 0; i < 9; ++i) { A[i] = SP[tid][i]; Bm[i] = SP[tid + s][i]; }
            #pragma unroll
            for (int i = 0; i < 3; ++i)
                #pragma unroll
                for (int j = 0; j < 3; ++j)
                    Cc[i * 3 + j] = lse3(A[i * 3 + 0] + Bm[0 * 3 + j],
                                         A[i * 3 + 1] + Bm[1 * 3 + j],
                                         A[i * 3 + 2] + Bm[2 * 3 + j]);
            #pragma unroll
            for (int i = 0; i < 9; ++i) SP[tid][i] = Cc[i];
            Snum[tid] += Snum[tid + s];
            Scnt[tid] += Scnt[tid + s];
        }
        __syncthreads();
    }

    if (tid == 0) {
        float a0 = st[0] + Ob[0], a1 = st[1] + Ob[1], a2 = st[2] + Ob[2];
        float f0 = lse3(a0 + SP[0][0], a1 + SP[0][3], a2 + SP[0][6]);
        float f1 = lse3(a0 + SP[0][1], a1 + SP[0][4], a2 + SP[0][7]);
        float f2 = lse3(a0 + SP[0][2], a1 + SP[0][5], a2 + SP[0][8]);
        float denom = lse3(f0 + et[0], f1 + et[1], f2 + et[2]);

        int total  = Scnt[0] + 1;           // mask[0] is forced true
        int ltag   = lab[total - 1];
        int l0     = lab[0];
        float num  = st[l0] + Ob[l0] + Snum[0] + et[ltag];
        unsafeAtomicAdd(out0, num - denom);
    }
}

// ------------------------------------------------------------------
extern "C" void kernel_launch(void* const* d_in, const int* in_sizes, int n_in,
                              void* d_out, int out_size, void* d_ws, size_t ws_size,
                              hipStream_t stream) {
    const float* x      = (const float*)d_in[0];   // (64,512,768)
    const int*   labels = (const int*)d_in[1];     // (64,512)
    const float* gamma  = (const float*)d_in[2];   // (768,)
    const float* beta   = (const float*)d_in[3];   // (768,)
    const float* W      = (const float*)d_in[4];   // (3,768)
    const float* bvec   = (const float*)d_in[5];   // (3,)
    const float* st     = (const float*)d_in[6];   // (3,)
    const float* et     = (const float*)d_in[7];   // (3,)
    const float* tr     = (const float*)d_in[8];   // (3,3)

    float* out  = (float*)d_out;        // out[0] = loglik, out[1..] = O flat
    float* ws   = (float*)d_ws;
    float* Omat = out + 1;

    k_init <<<1,   256, 0, stream>>>(ws, out);
    k_stats<<<256, 256, 0, stream>>>(x, ws);
    k_prep <<<1,   256, 0, stream>>>(gamma, beta, W, bvec, ws);
    k_gemm <<<256, 256, 0, stream>>>(x, ws, Omat);
    k_crf  <<<Bn,  128, 0, stream>>>(Omat, labels, st, et, tr, out);
}